// Detector_5274219839804
// MI455X (gfx1250) — compile-verified
//
#include <hip/hip_runtime.h>
#include <hip/hip_bf16.h>
#include <math.h>

typedef unsigned long long u64;
typedef __attribute__((ext_vector_type(2))) float v2f;
typedef __attribute__((ext_vector_type(8))) float v8f;

#define NPTS 1024   // points per submap
#define KPOS 512    // closest points kept
#define KSEL 32     // FPS samples

__global__ __launch_bounds__(1024, 1)
void detector_fused(const float* __restrict__ x,
                    const float* __restrict__ pos,
                    const float* __restrict__ W1, const float* __restrict__ b1,
                    const float* __restrict__ W2, const float* __restrict__ b2,
                    const float* __restrict__ W3, const float* __restrict__ b3,
                    float* __restrict__ out, int B)
{
    __shared__ u64   keys[NPTS];        // (d2_bits << 32) | idx
    __shared__ int   idxsel[KPOS];
    __shared__ float px[KPOS], py[KPOS], pz[KPOS];
    __shared__ float mind[KPOS];
    __shared__ u64   red[KPOS];
    __shared__ int   sel[KSEL];
    __shared__ int   orig[KSEL];
    __shared__ float xa[KSEL * 32];     // gathered x rows
    __shared__ float h1[KSEL * 16];     // layer-1 activations

    const int tid = threadIdx.x;
    const int b   = blockIdx.x;

    // ---- 1) xy-distance^2 sort key (monotone float bits, index tie-break) ----
    {
        size_t base = ((size_t)b * NPTS + tid) * 3;
        float X = pos[base + 0];
        float Y = pos[base + 1];
        float d2 = X * X + Y * Y;                       // >= 0 -> bits monotone
        keys[tid] = ((u64)__float_as_uint(d2) << 32) | (u64)tid;
    }
    __syncthreads();

    // ---- 2) bitonic sort ascending (1024 elems, 1 thread / elem) ----
    for (int k = 2; k <= NPTS; k <<= 1) {
        for (int j = k >> 1; j > 0; j >>= 1) {
            int ixj = tid ^ j;
            if (ixj > tid) {
                u64 a = keys[tid];
                u64 c = keys[ixj];
                bool up = ((tid & k) == 0);
                if (up ? (a > c) : (a < c)) { keys[tid] = c; keys[ixj] = a; }
            }
            __syncthreads();
        }
    }

    // ---- 3) take 512 closest, gather their 3D coords into LDS ----
    if (tid < KPOS) {
        int idx = (int)(keys[tid] & 0xFFFFFFFFull);
        idxsel[tid] = idx;
        size_t base = ((size_t)b * NPTS + idx) * 3;
        px[tid] = pos[base + 0];
        py[tid] = pos[base + 1];
        pz[tid] = pos[base + 2];
    }
    if (tid == 0) sel[0] = 0;    // FPS starts at the closest point
    __syncthreads();

    // ---- 4) FPS: init min-dist to point 0 ----
    if (tid < KPOS) {
        float dx = px[tid] - px[0];
        float dy = py[tid] - py[0];
        float dz = pz[tid] - pz[0];
        mind[tid] = dx*dx + dy*dy + dz*dz;
    }
    __syncthreads();

    for (int t = 1; t < KSEL; ++t) {
        // argmax(mind) with first-index tie-break: pack (bits, 511-tid)
        if (tid < KPOS)
            red[tid] = ((u64)__float_as_uint(mind[tid]) << 32) | (u64)(KPOS - 1 - tid);
        __syncthreads();
        for (int s = KPOS / 2; s > 0; s >>= 1) {
            if (tid < s) {
                u64 a = red[tid], c = red[tid + s];
                red[tid] = (a > c) ? a : c;
            }
            __syncthreads();
        }
        int j = KPOS - 1 - (int)(red[0] & 0xFFFFFFFFull);
        if (tid == 0) sel[t] = j;
        if (tid < KPOS) {
            float dx = px[tid] - px[j];
            float dy = py[tid] - py[j];
            float dz = pz[tid] - pz[j];
            float d2 = dx*dx + dy*dy + dz*dz;
            mind[tid] = fminf(mind[tid], d2);
        }
        __syncthreads();
    }

    // ---- 5) map to original within-submap indices ----
    if (tid < KSEL) orig[tid] = idxsel[sel[tid]];
    __syncthreads();

    // ---- 6) gather x rows (32 rows x 32 feats, fully coalesced) ----
    {
        int r = tid >> 5;
        int c = tid & 31;
        xa[tid] = x[((size_t)b * NPTS + (size_t)orig[r]) * 32 + c];
    }
    __syncthreads();

    // ---- 7) layer 1: h1 = relu(xa @ W1 + b1) via v_wmma_f32_16x16x4_f32 ----
    // waves 0 and 1 each own a 16-row tile; 8 WMMAs accumulate K=32.
    if (tid < 64) {
        const int lane    = tid & 31;
        const int rowbase = (tid >> 5) * 16;
        const int m       = lane & 15;          // row (A) / col N (B,C)
        const int khalf   = (lane >> 4) * 2;    // K sub-offset per half-wave
        v8f acc = {};
        #pragma unroll
        for (int c = 0; c < 8; ++c) {
            v2f a, w;
            // A 16x4 fp32 layout: lanes 0-15 hold K=0,1 ; lanes 16-31 hold K=2,3
            a.x = xa[(rowbase + m) * 32 + 4 * c + khalf];
            a.y = xa[(rowbase + m) * 32 + 4 * c + khalf + 1];
            // B 4x16 fp32 layout: lanes 0-15 rows K=0,1 ; lanes 16-31 rows K=2,3
            int k0 = 4 * c + khalf;
            w.x = W1[(size_t)k0 * 16 + m];
            w.y = W1[(size_t)(k0 + 1) * 16 + m];
            acc = __builtin_amdgcn_wmma_f32_16x16x4_f32(
                false, a, false, w, (short)0, acc, false, false);
        }
        // C/D layout: VGPR r -> row r (lanes 0-15) / row r+8 (lanes 16-31), col = lane&15
        float bias = b1[m];
        #pragma unroll
        for (int r = 0; r < 8; ++r) {
            int row = rowbase + r + ((lane >> 4) ? 8 : 0);
            float v = acc[r] + bias;
            h1[row * 16 + m] = v > 0.0f ? v : 0.0f;
        }
    }
    __syncthreads();

    // ---- 8) layers 2-3 + softplus, emit outputs ----
    if (tid < KSEL) {
        float s = b3[0];
        #pragma unroll
        for (int jj = 0; jj < 8; ++jj) {
            float h = b2[jj];
            #pragma unroll
            for (int i = 0; i < 16; ++i)
                h += h1[tid * 16 + i] * W2[i * 8 + jj];
            h = h > 0.0f ? h : 0.0f;
            s += h * W3[jj];
        }
        float sp = (s > 20.0f) ? s : log1pf(expf(s));   // stable softplus
        out[(size_t)b * KSEL + tid] = sp;                               // weights
        out[(size_t)B * KSEL + (size_t)b * KSEL + tid] = (float)orig[tid]; // indices
    }
}

extern "C" void kernel_launch(void* const* d_in, const int* in_sizes, int n_in,
                              void* d_out, int out_size, void* d_ws, size_t ws_size,
                              hipStream_t stream) {
    const float* x   = (const float*)d_in[0];
    const float* pos = (const float*)d_in[1];
    // d_in[2] = batch (int32) unused: layout is fixed B x 1024
    const float* W1  = (const float*)d_in[3];
    const float* b1  = (const float*)d_in[4];
    const float* W2  = (const float*)d_in[5];
    const float* b2  = (const float*)d_in[6];
    const float* W3  = (const float*)d_in[7];
    const float* b3  = (const float*)d_in[8];

    int B = in_sizes[2] / NPTS;   // batch vector has B*1024 entries

    detector_fused<<<dim3(B), dim3(NPTS), 0, stream>>>(
        x, pos, W1, b1, W2, b2, W3, b3, (float*)d_out, B);
}